// GSRS_69836168233152
// MI455X (gfx1250) — compile-verified
//
#include <hip/hip_runtime.h>
#include <stdint.h>

// Problem constants from the reference.
#define NB 32
#define NC 512
#define NH 256
#define NW 14
#define HW (NH * NW)        // 3584 floats per (b,c) plane = 14336 bytes (16B aligned)
#define CHUNKS (HW / 4)     // 896 float4 (16B) chunks per plane
#define THREADS 256

// One block per (b,c) plane. The (H,W) plane is contiguous in memory, so both
// the global load (async -> LDS, 16B per lane) and the global store (float4)
// are fully coalesced; the awkward 14-wide in-row shuffle is resolved as an
// LDS gather using a per-row map shared across all channels of a batch.
__global__ __launch_bounds__(THREADS)
void GSRS_69836168233152_kernel(const float* __restrict__ x,
                                const int* __restrict__ perms,
                                float* __restrict__ out) {
    __shared__ float tile[HW];             // 14336 B staged x plane
    __shared__ unsigned short map16[HW];   // 7168 B: absolute LDS source index per element

    const int tid = threadIdx.x;
    const int blk = blockIdx.x;            // [0, NB*NC)
    const int b = blk >> 9;                // / NC (NC == 512)
    const int c = blk & (NC - 1);

    const float* __restrict__ gsrc = x + ((size_t)b * NC + c) * HW;
    float* __restrict__ gdst = out + ((size_t)b * NC + c) * HW;
    const int* __restrict__ prow = perms + (size_t)b * NH * 7;

    // ---- Issue async global->LDS copies of the x plane (CDNA5 async path). ----
    // Per-lane 16B transfer: LDS byte offset in a VGPR, 64-bit global address pair.
    const unsigned lds_base = (unsigned)(uintptr_t)(&tile[0]);
#pragma unroll
    for (int it = 0; it < 4; ++it) {
        int ch = tid + it * THREADS;
        if (ch < CHUNKS) {
            unsigned laddr = lds_base + (unsigned)(ch * 16);
            const float* g = gsrc + ch * 4;
            asm volatile("global_load_async_to_lds_b128 %0, %1, off"
                         :: "v"(laddr), "v"(g)
                         : "memory");
        }
    }

    // ---- Overlap: build the per-row column remap while the DMA is in flight.
    // Row h (parity p = h&1): columns with (w&1)==p are shuffled slots,
    // slot j = (w-p)/2 sources column 2*perms[b,h,j]+p; others are identity.
    for (int f = tid; f < HW; f += THREADS) {
        int h = f / NW;
        int w = f - h * NW;
        int parity = h & 1;
        int src;
        if ((w & 1) == parity) {
            int j = (w - parity) >> 1;
            src = 2 * prow[h * 7 + j] + parity;
        } else {
            src = w;
        }
        map16[f] = (unsigned short)(h * NW + src);
    }

    // Wait for this wave's async transfers, then make LDS visible block-wide.
    asm volatile("s_wait_asynccnt 0" ::: "memory");
    __syncthreads();

    // ---- Gather through LDS and store fully coalesced 16B per lane. ----
#pragma unroll
    for (int it = 0; it < 4; ++it) {
        int ch = tid + it * THREADS;
        if (ch < CHUNKS) {
            int e = ch * 4;
            float4 v;
            v.x = tile[map16[e + 0]];
            v.y = tile[map16[e + 1]];
            v.z = tile[map16[e + 2]];
            v.w = tile[map16[e + 3]];
            reinterpret_cast<float4*>(gdst)[ch] = v;
        }
    }
}

extern "C" void kernel_launch(void* const* d_in, const int* in_sizes, int n_in,
                              void* d_out, int out_size, void* d_ws, size_t ws_size,
                              hipStream_t stream) {
    const float* x = (const float*)d_in[0];       // [B,C,H,W] float32
    const int* perms = (const int*)d_in[1];       // [B,H,7] int32
    float* out = (float*)d_out;                   // [B,C,H,W] float32

    dim3 grid(NB * NC);   // 16384 blocks, one per (b,c) plane
    GSRS_69836168233152_kernel<<<grid, THREADS, 0, stream>>>(x, perms, out);
}